// GlobalPointer_78271484003185
// MI455X (gfx1250) — compile-verified
//
#include <hip/hip_runtime.h>

typedef __attribute__((ext_vector_type(2))) float v2f;
typedef __attribute__((ext_vector_type(8))) float v8f;

#define ENT    16
#define INNER  64
#define NEGV   1000000000000.0f
#define S_LEN  1024
#define H_DIM  1024
#define N_OUT  2048   // ENT * INNER * 2
#define B_SZ   4

// ---------------------------------------------------------------------------
// Kernel 1: (B*S, H) @ (H, 2*ENT*INNER) + bias, then interleaved RoPE,
// scatter into Qws/Kws laid out [B, ENT, S, INNER].
// Block = 32x8 (8 waves). Each wave computes a 16(M) x 64(N) strip with
// 4 fp32 WMMA accumulators (V_WMMA_F32_16X16X4_F32 over K=1024).
// ---------------------------------------------------------------------------
__global__ __launch_bounds__(256)
void gp_qk_rope_kernel(const float* __restrict__ X, const float* __restrict__ W,
                       const float* __restrict__ bias,
                       float* __restrict__ Qws, float* __restrict__ Kws)
{
    const int lane = threadIdx.x;          // 0..31 (wave32)
    const int wave = threadIdx.y;          // 0..7
    const int hf   = lane >> 4;            // lane half: K sub-block selector
    const int l15  = lane & 15;

    const int mBase = blockIdx.x * 128 + wave * 16;   // row in (B*S)
    const int nBase = blockIdx.y * 64;                // col in N_OUT

    v8f acc[4] = {};
    for (int kb = 0; kb < H_DIM; kb += 4) {
        const int k0 = kb + hf * 2;
        // A frag: 16x4 fp32 — lane l15 holds row (mBase+l15), K = k0, k0+1
        v2f afrag = *(const v2f*)(X + (size_t)(mBase + l15) * H_DIM + k0);
#pragma unroll
        for (int t = 0; t < 4; ++t) {
            const int col = nBase + t * 16 + l15;
            v2f bfrag;
            bfrag.x = W[(size_t)k0 * N_OUT + col];
            bfrag.y = W[(size_t)(k0 + 1) * N_OUT + col];
            acc[t] = __builtin_amdgcn_wmma_f32_16x16x4_f32(
                false, afrag, false, bfrag, (short)0, acc[t], false, false);
        }
    }

    // Epilogue: bias + interleaved RoPE. Lane owns column `col` for 8 rows;
    // the rotate-half partner (col^1) lives in lane^1 -> one shfl_xor.
#pragma unroll
    for (int t = 0; t < 4; ++t) {
        const int col = nBase + t * 16 + l15;
        const int h   = col >> 7;          // head slot (0..15)
        const int dq  = col & 127;         // 0..63 -> q, 64..127 -> k
        const int d   = dq & 63;           // inner dim index
        const int j   = d >> 1;
        const float freq = __powf(10000.0f, -(float)(2 * j) * (1.0f / 64.0f));
        const float bval = bias[col];
#pragma unroll
        for (int i = 0; i < 8; ++i) {
            const int row = mBase + hf * 8 + i;
            const int s   = row & (S_LEN - 1);
            const int bb  = row >> 10;
            float v = acc[t][i] + bval;
            float partner = __shfl_xor(v, 1, 32);   // value at column col^1
            float sn, cs;
            __sincosf((float)s * freq, &sn, &cs);
            const float rot = (d & 1) ? partner : -partner;
            const float outv = v * cs + rot * sn;
            float* dst = (dq < INNER) ? Qws : Kws;
            dst[(((size_t)bb * ENT + h) * S_LEN + s) * INNER + d] = outv;
        }
    }
}

// ---------------------------------------------------------------------------
// Kernel 2: per (b,h): logits = Q(1024x64) @ K^T(64x1024), pad mask,
// strict-lower-tri causal mask, * 1/sqrt(INNER).
// Block = 32x8; each wave computes a 16(M) x 128(N) strip (8 accumulators).
// Grid: (S/128, S/128, B*ENT).
// ---------------------------------------------------------------------------
__global__ __launch_bounds__(256)
void gp_logits_kernel(const float* __restrict__ Qws, const float* __restrict__ Kws,
                      const float* __restrict__ tok, float* __restrict__ out)
{
    const int lane = threadIdx.x;
    const int wave = threadIdx.y;
    const int hf   = lane >> 4;
    const int l15  = lane & 15;

    const int bh    = blockIdx.z;                    // b*ENT + h
    const int b     = bh >> 4;
    const int mBase = blockIdx.y * 128 + wave * 16;
    const int nBase = blockIdx.x * 128;

    const float* Q = Qws + (size_t)bh * S_LEN * INNER;
    const float* K = Kws + (size_t)bh * S_LEN * INNER;

    v8f acc[8] = {};
    for (int kb = 0; kb < INNER; kb += 4) {
        const int k0 = kb + hf * 2;
        v2f afrag = *(const v2f*)(Q + (size_t)(mBase + l15) * INNER + k0);
#pragma unroll
        for (int t = 0; t < 8; ++t) {
            // B = K^T : B[k][n] = K[n*64 + k] -> contiguous float2 per lane
            v2f bfrag = *(const v2f*)(K + (size_t)(nBase + t * 16 + l15) * INNER + k0);
            acc[t] = __builtin_amdgcn_wmma_f32_16x16x4_f32(
                false, afrag, false, bfrag, (short)0, acc[t], false, false);
        }
    }

#pragma unroll
    for (int t = 0; t < 8; ++t) {
        const int n = nBase + t * 16 + l15;
        const float pad = tok[b * S_LEN + n];
        const float padTerm = (1.0f - pad) * NEGV;
#pragma unroll
        for (int i = 0; i < 8; ++i) {
            const int m = mBase + hf * 8 + i;
            float v = acc[t][i] * pad - padTerm;
            if (m > n) v -= NEGV;                    // tril(ones, -1) mask
            v *= 0.125f;                             // 1/sqrt(64)
            out[((size_t)bh * S_LEN + m) * S_LEN + n] = v;
        }
    }
}

// ---------------------------------------------------------------------------
extern "C" void kernel_launch(void* const* d_in, const int* in_sizes, int n_in,
                              void* d_out, int out_size, void* d_ws, size_t ws_size,
                              hipStream_t stream)
{
    const float* X    = (const float*)d_in[0];   // (4,1024,1024)
    const float* W    = (const float*)d_in[1];   // (1024,2048)
    const float* bias = (const float*)d_in[2];   // (2048,)
    const float* tok  = (const float*)d_in[3];   // (4,1024)

    float* Qws = (float*)d_ws;                                   // 16 MB
    float* Kws = Qws + (size_t)B_SZ * ENT * S_LEN * INNER;       // +16 MB
    float* out = (float*)d_out;                                  // (4,16,1024,1024)

    dim3 blk(32, 8);
    dim3 g1((B_SZ * S_LEN) / 128, N_OUT / 64);   // (32, 32)
    gp_qk_rope_kernel<<<g1, blk, 0, stream>>>(X, W, bias, Qws, Kws);

    dim3 g2(S_LEN / 128, S_LEN / 128, B_SZ * ENT);  // (8, 8, 64)
    gp_logits_kernel<<<g2, blk, 0, stream>>>(Qws, Kws, tok, out);
}